// DiffusionModel_76063870812655
// MI455X (gfx1250) — compile-verified
//
#include <hip/hip_runtime.h>

typedef float v2f __attribute__((ext_vector_type(2)));
typedef float v8f __attribute__((ext_vector_type(8)));

#define N_USER 100000
#define N_ITEM 50000
#define NEDGE  1600000
#define HDIM   256

// ---------------------------------------------------------------------------
// Scatter: one wave32 per edge. Lanes stripe the K feature columns and
// atomically accumulate the source row into Acc[dst]; lane 0 bumps the count.
// ---------------------------------------------------------------------------
__global__ __launch_bounds__(256) void sage_scatter(
    const float* __restrict__ X, const int* __restrict__ src,
    const int* __restrict__ dst, float* __restrict__ Acc,
    float* __restrict__ Cnt, int E, int K)
{
    int wid  = (int)((blockIdx.x * blockDim.x + threadIdx.x) >> 5);
    int lane = threadIdx.x & 31;
    if (wid >= E) return;
    int s = src[wid];
    int d = dst[wid];
    const float* xr = X + (size_t)s * K;
    float* ar = Acc + (size_t)d * K;
    for (int c = lane; c < K; c += 32)
        atomicAdd(ar + c, xr[c]);
    if (lane == 0) atomicAdd(Cnt + d, 1.0f);
}

// cnt -> 1/cnt (0 for isolated nodes), in place
__global__ __launch_bounds__(256) void sage_inv(float* __restrict__ Cnt, int N)
{
    int i = blockIdx.x * blockDim.x + threadIdx.x;
    if (i < N) {
        float c = Cnt[i];
        Cnt[i] = (c > 0.0f) ? (1.0f / c) : 0.0f;
    }
}

// ---------------------------------------------------------------------------
// Fused SAGE output GEMM: Out = (Acc*Inv) @ Wl^T + Bias + Xd @ Wr^T, opt ReLU.
// One wave per 16(M) x 64(N) tile; 4 WMMA col-tiles share each A fragment.
// A tile staged in LDS with padded stride 260 to dodge bank conflicts.
// WMMA fp32 16x16x4 fragment layout (ISA 7.12.2):
//   A: lanes 0-15 row M=l16 hold K=k0,k0+1; lanes 16-31 hold K=k0+2,k0+3
//   B: v0/v1 hold rows K (=W^T rows) striped the same way, N = l16
//   C/D: vgpr v -> M = v + 8*half, N = l16
// ---------------------------------------------------------------------------
template <int KA, int KR, int RELU>
__global__ __launch_bounds__(32) void sage_gemm(
    const float* __restrict__ Acc, const float* __restrict__ Inv,
    const float* __restrict__ Wl,  const float* __restrict__ Bias,
    const float* __restrict__ Xd,  const float* __restrict__ Wr,
    float* __restrict__ Out, int M)
{
    constexpr int LDT = HDIM + 4;          // 260-float row stride (bank-safe)
    __shared__ float tile[16 * LDT];
    const int lane = threadIdx.x;
    const int hi   = lane >> 4;            // wave half
    const int l16  = lane & 15;
    const int m0   = blockIdx.x * 16;
    const int n0   = blockIdx.y * 64;

    v8f c0 = {}, c1 = {}, c2 = {}, c3 = {};

    // ---------------- term 1: mean(Acc*Inv) @ Wl^T (K = KA) ----------------
    for (int idx = lane; idx < 4 * KA; idx += 32) {
        int flat = idx * 4;
        int r    = flat / KA;
        int col  = flat - r * KA;
        float4 v = *(const float4*)(Acc + (size_t)(m0 + r) * KA + col);
        float  s = Inv[m0 + r];
        float* t = tile + r * LDT + col;
        t[0] = v.x * s; t[1] = v.y * s; t[2] = v.z * s; t[3] = v.w * s;
    }
    __syncthreads();
    for (int k0 = 0; k0 < KA; k0 += 4) {
        v2f a  = *(const v2f*)(tile + l16 * LDT + k0 + 2 * hi);
        v2f b0 = *(const v2f*)(Wl + (size_t)(n0 +  0 + l16) * KA + k0 + 2 * hi);
        v2f b1 = *(const v2f*)(Wl + (size_t)(n0 + 16 + l16) * KA + k0 + 2 * hi);
        v2f b2 = *(const v2f*)(Wl + (size_t)(n0 + 32 + l16) * KA + k0 + 2 * hi);
        v2f b3 = *(const v2f*)(Wl + (size_t)(n0 + 48 + l16) * KA + k0 + 2 * hi);
        c0 = __builtin_amdgcn_wmma_f32_16x16x4_f32(false, a, false, b0, (short)0, c0, false, false);
        c1 = __builtin_amdgcn_wmma_f32_16x16x4_f32(false, a, false, b1, (short)0, c1, false, false);
        c2 = __builtin_amdgcn_wmma_f32_16x16x4_f32(false, a, false, b2, (short)0, c2, false, false);
        c3 = __builtin_amdgcn_wmma_f32_16x16x4_f32(false, a, false, b3, (short)0, c3, false, false);
    }
    __syncthreads();

    // ---------------- term 2: Xd @ Wr^T (K = KR) ----------------
    for (int idx = lane; idx < 4 * KR; idx += 32) {
        int flat = idx * 4;
        int r    = flat / KR;
        int col  = flat - r * KR;
        float4 v = *(const float4*)(Xd + (size_t)(m0 + r) * KR + col);
        float* t = tile + r * LDT + col;
        t[0] = v.x; t[1] = v.y; t[2] = v.z; t[3] = v.w;
    }
    __syncthreads();
    for (int k0 = 0; k0 < KR; k0 += 4) {
        v2f a  = *(const v2f*)(tile + l16 * LDT + k0 + 2 * hi);
        v2f b0 = *(const v2f*)(Wr + (size_t)(n0 +  0 + l16) * KR + k0 + 2 * hi);
        v2f b1 = *(const v2f*)(Wr + (size_t)(n0 + 16 + l16) * KR + k0 + 2 * hi);
        v2f b2 = *(const v2f*)(Wr + (size_t)(n0 + 32 + l16) * KR + k0 + 2 * hi);
        v2f b3 = *(const v2f*)(Wr + (size_t)(n0 + 48 + l16) * KR + k0 + 2 * hi);
        c0 = __builtin_amdgcn_wmma_f32_16x16x4_f32(false, a, false, b0, (short)0, c0, false, false);
        c1 = __builtin_amdgcn_wmma_f32_16x16x4_f32(false, a, false, b1, (short)0, c1, false, false);
        c2 = __builtin_amdgcn_wmma_f32_16x16x4_f32(false, a, false, b2, (short)0, c2, false, false);
        c3 = __builtin_amdgcn_wmma_f32_16x16x4_f32(false, a, false, b3, (short)0, c3, false, false);
    }

    // ---------------- bias + optional ReLU + store ----------------
    const float bb0 = Bias[n0 +  0 + l16];
    const float bb1 = Bias[n0 + 16 + l16];
    const float bb2 = Bias[n0 + 32 + l16];
    const float bb3 = Bias[n0 + 48 + l16];
#pragma unroll
    for (int v = 0; v < 8; ++v) {
        int row = m0 + v + 8 * hi;
        float* orow = Out + (size_t)row * HDIM + n0;
        float o0 = c0[v] + bb0;
        float o1 = c1[v] + bb1;
        float o2 = c2[v] + bb2;
        float o3 = c3[v] + bb3;
        if (RELU) {
            o0 = fmaxf(o0, 0.0f); o1 = fmaxf(o1, 0.0f);
            o2 = fmaxf(o2, 0.0f); o3 = fmaxf(o3, 0.0f);
        }
        orow[ 0 + l16] = o0;
        orow[16 + l16] = o1;
        orow[32 + l16] = o2;
        orow[48 + l16] = o3;
    }
}

extern "C" void kernel_launch(void* const* d_in, const int* in_sizes, int n_in,
                              void* d_out, int out_size, void* d_ws, size_t ws_size,
                              hipStream_t stream)
{
    (void)in_sizes; (void)n_in; (void)out_size; (void)ws_size;

    const float* x_user  = (const float*)d_in[0];
    const float* x_item  = (const float*)d_in[1];
    const int*   e_ui_s  = (const int*)d_in[2];
    const int*   e_ui_d  = (const int*)d_in[3];
    const int*   e_iu_s  = (const int*)d_in[4];
    const int*   e_iu_d  = (const int*)d_in[5];
    const float* W1l_ui  = (const float*)d_in[6];
    const float* b1_ui   = (const float*)d_in[7];
    const float* W1r_ui  = (const float*)d_in[8];
    const float* W1l_iu  = (const float*)d_in[9];
    const float* b1_iu   = (const float*)d_in[10];
    const float* W1r_iu  = (const float*)d_in[11];
    const float* W2l_ui  = (const float*)d_in[12];
    const float* b2_ui   = (const float*)d_in[13];
    const float* W2r_ui  = (const float*)d_in[14];
    const float* W2l_iu  = (const float*)d_in[15];
    const float* b2_iu   = (const float*)d_in[16];
    const float* W2r_iu  = (const float*)d_in[17];

    // workspace layout (floats): h_user | h_item | acc (reused) | cnt
    float* h_user = (float*)d_ws;
    float* h_item = h_user + (size_t)N_USER * HDIM;
    float* acc    = h_item + (size_t)N_ITEM * HDIM;
    float* cnt    = acc    + (size_t)N_USER * HDIM;

    float* out_user = (float*)d_out;
    float* out_item = out_user + (size_t)N_USER * HDIM;

    const int scat_blocks = NEDGE / 8;   // 8 waves (edges) per 256-thread block

    // ---- layer 1, user -> item : h_item = relu(conv) ----
    hipMemsetAsync(acc, 0, (size_t)N_ITEM * 128 * sizeof(float), stream);
    hipMemsetAsync(cnt, 0, (size_t)N_ITEM * sizeof(float), stream);
    sage_scatter<<<scat_blocks, 256, 0, stream>>>(x_user, e_ui_s, e_ui_d, acc, cnt, NEDGE, 128);
    sage_inv<<<(N_ITEM + 255) / 256, 256, 0, stream>>>(cnt, N_ITEM);
    sage_gemm<128, 64, 1><<<dim3(N_ITEM / 16, HDIM / 64), 32, 0, stream>>>(
        acc, cnt, W1l_ui, b1_ui, x_item, W1r_ui, h_item, N_ITEM);

    // ---- layer 1, item -> user : h_user = relu(conv) ----
    hipMemsetAsync(acc, 0, (size_t)N_USER * 64 * sizeof(float), stream);
    hipMemsetAsync(cnt, 0, (size_t)N_USER * sizeof(float), stream);
    sage_scatter<<<scat_blocks, 256, 0, stream>>>(x_item, e_iu_s, e_iu_d, acc, cnt, NEDGE, 64);
    sage_inv<<<(N_USER + 255) / 256, 256, 0, stream>>>(cnt, N_USER);
    sage_gemm<64, 128, 1><<<dim3(N_USER / 16, HDIM / 64), 32, 0, stream>>>(
        acc, cnt, W1l_iu, b1_iu, x_user, W1r_iu, h_user, N_USER);

    // ---- layer 2, user -> item : out_item ----
    hipMemsetAsync(acc, 0, (size_t)N_ITEM * HDIM * sizeof(float), stream);
    hipMemsetAsync(cnt, 0, (size_t)N_ITEM * sizeof(float), stream);
    sage_scatter<<<scat_blocks, 256, 0, stream>>>(h_user, e_ui_s, e_ui_d, acc, cnt, NEDGE, HDIM);
    sage_inv<<<(N_ITEM + 255) / 256, 256, 0, stream>>>(cnt, N_ITEM);
    sage_gemm<256, 256, 0><<<dim3(N_ITEM / 16, HDIM / 64), 32, 0, stream>>>(
        acc, cnt, W2l_ui, b2_ui, h_item, W2r_ui, out_item, N_ITEM);

    // ---- layer 2, item -> user : out_user ----
    hipMemsetAsync(acc, 0, (size_t)N_USER * HDIM * sizeof(float), stream);
    hipMemsetAsync(cnt, 0, (size_t)N_USER * sizeof(float), stream);
    sage_scatter<<<scat_blocks, 256, 0, stream>>>(h_item, e_iu_s, e_iu_d, acc, cnt, NEDGE, HDIM);
    sage_inv<<<(N_USER + 255) / 256, 256, 0, stream>>>(cnt, N_USER);
    sage_gemm<256, 256, 0><<<dim3(N_USER / 16, HDIM / 64), 32, 0, stream>>>(
        acc, cnt, W2l_iu, b2_iu, h_user, W2r_iu, out_user, N_USER);
}